// AttentionMechanism_55061480735373
// MI455X (gfx1250) — compile-verified
//
#include <hip/hip_runtime.h>
#include <math.h>

typedef __attribute__((ext_vector_type(2))) float v2f;
typedef __attribute__((ext_vector_type(8))) float v8f;

// ---------------------------------------------------------------------------
// Kernel 1: v[i] = sum_o a[64+o] * W[o,i]   (the a^T-half folded into W)
// ---------------------------------------------------------------------------
__global__ void k_fold_v(const float* __restrict__ W, const float* __restrict__ a,
                         float* __restrict__ v) {
    int i = threadIdx.x;              // 64 threads
    float acc = 0.f;
#pragma unroll 8
    for (int o = 0; o < 64; ++o) acc += a[64 + o] * W[o * 64 + i];
    v[i] = acc;
}

// ---------------------------------------------------------------------------
// Kernel 2: s[j] = sum_i v[i] * flat[i*N + j]  (weighted sum of 64 segments,
// float4 streaming, 4096 floats per block) + per-block max for softmax.
// Full blocks take a guard-free path; only the tail block checks bounds.
// ---------------------------------------------------------------------------
__global__ void k_energy(const float4* __restrict__ flat4, const float* __restrict__ v,
                         float4* __restrict__ s4, float* __restrict__ bmax, int N4) {
    __shared__ float sv[64];
    __shared__ float smax[256];
    int tid = threadIdx.x;
    if (tid < 64) sv[tid] = v[tid];
    __syncthreads();

    int blockBase4 = blockIdx.x * 1024;          // float4 units
    int base4 = blockBase4 + tid;
    float4 acc[4];
#pragma unroll
    for (int t = 0; t < 4; ++t) acc[t] = make_float4(0.f, 0.f, 0.f, 0.f);

    if (blockBase4 + 1024 <= N4) {
        // ---- fast path: no bounds checks ----
        for (int i = 0; i < 64; ++i) {
            float vi = sv[i];
            const float4* seg = flat4 + (size_t)i * (size_t)N4 + base4;
#pragma unroll
            for (int t = 0; t < 4; ++t) {
                float4 d = seg[t * 256];
                acc[t].x += vi * d.x; acc[t].y += vi * d.y;
                acc[t].z += vi * d.z; acc[t].w += vi * d.w;
            }
        }
        float m = -INFINITY;
#pragma unroll
        for (int t = 0; t < 4; ++t) {
            s4[base4 + t * 256] = acc[t];
            m = fmaxf(m, fmaxf(fmaxf(acc[t].x, acc[t].y), fmaxf(acc[t].z, acc[t].w)));
        }
        smax[tid] = m;
    } else {
        // ---- tail block ----
        for (int i = 0; i < 64; ++i) {
            float vi = sv[i];
            const float4* seg = flat4 + (size_t)i * (size_t)N4;
#pragma unroll
            for (int t = 0; t < 4; ++t) {
                int f = base4 + t * 256;
                if (f < N4) {
                    float4 d = seg[f];
                    acc[t].x += vi * d.x; acc[t].y += vi * d.y;
                    acc[t].z += vi * d.z; acc[t].w += vi * d.w;
                }
            }
        }
        float m = -INFINITY;
#pragma unroll
        for (int t = 0; t < 4; ++t) {
            int f = base4 + t * 256;
            if (f < N4) {
                s4[f] = acc[t];
                m = fmaxf(m, fmaxf(fmaxf(acc[t].x, acc[t].y), fmaxf(acc[t].z, acc[t].w)));
            }
        }
        smax[tid] = m;
    }
    __syncthreads();
    for (int w = 128; w > 0; w >>= 1) {
        if (tid < w) smax[tid] = fmaxf(smax[tid], smax[tid + w]);
        __syncthreads();
    }
    if (tid == 0) bmax[blockIdx.x] = smax[0];
}

// ---------------------------------------------------------------------------
// Kernel 3: global max over block maxima
// ---------------------------------------------------------------------------
__global__ void k_max(const float* __restrict__ bmax, float* __restrict__ mOut, int nb) {
    __shared__ float sm[256];
    int tid = threadIdx.x;
    float m = -INFINITY;
    for (int i = tid; i < nb; i += 256) m = fmaxf(m, bmax[i]);
    sm[tid] = m;
    __syncthreads();
    for (int w = 128; w > 0; w >>= 1) {
        if (tid < w) sm[tid] = fmaxf(sm[tid], sm[tid + w]);
        __syncthreads();
    }
    if (tid == 0) mOut[0] = sm[0];
}

// ---------------------------------------------------------------------------
// Kernel 4: per-block partial Z = sum exp(s-m) and partial agg[0..63] via
// V_WMMA_F32_16X16X4_F32.  A(16x4) = exp-weights replicated over M rows,
// B(4x16) = n_x rows; every C row holds the identical partial sum -> row M=0.
// Block = 256 threads = 8 waves, 4096 neighbors per block (512 per wave).
// Fast path for full blocks: pointer-marching, zero guards, EXEC all-1s.
// ---------------------------------------------------------------------------
__global__ void k_agg(const float* __restrict__ nx, const float* __restrict__ s,
                      const float* __restrict__ mPtr, float* __restrict__ bsum,
                      float* __restrict__ bagg, int N) {
    __shared__ float lds_agg[8 * 64];
    __shared__ float lds_z[256];
    int tid = threadIdx.x;
    float m = mPtr[0];
    int blockBase = blockIdx.x * 4096;
    bool full = (blockBase + 4096 <= N);

    // --- partial Z (deterministic fixed-order block reduce) ---
    float z = 0.f;
    if (full) {
#pragma unroll
        for (int t = 0; t < 16; ++t) z += __expf(s[blockBase + tid + t * 256] - m);
    } else {
#pragma unroll
        for (int t = 0; t < 16; ++t) {
            int j = blockBase + tid + t * 256;
            float e = __expf(s[min(j, N - 1)] - m);
            z += (j < N) ? e : 0.f;               // select, not branch
        }
    }
    lds_z[tid] = z;
    __syncthreads();
    for (int w = 128; w > 0; w >>= 1) {
        if (tid < w) lds_z[tid] += lds_z[tid + w];
        __syncthreads();
    }
    if (tid == 0) bsum[blockIdx.x] = lds_z[0];

    // --- WMMA partial agg; EXEC is all-1s at every WMMA ---
    int wave = tid >> 5;
    int lane = tid & 31;
    int hi   = (lane >= 16) ? 1 : 0;              // which K-half this lane holds
    int col  = lane - (hi ? 16 : 0);
    int jbase = blockBase + wave * 512;

    v8f c0 = {}, c1 = {}, c2 = {}, c3 = {};

    if (full) {
        // rows b0 = jbase+4*stp+(hi?2:0), b1 = b0+1 are consecutive:
        // all 8 B values come off one marching pointer with imm offsets.
        const float* p  = nx + (size_t)(jbase + (hi ? 2 : 0)) * 64 + (size_t)col;
        const float* sp = s + jbase + (hi ? 2 : 0);
        for (int stp = 0; stp < 128; ++stp) {
            v2f sv = *(const v2f*)sp;             // s[b0], s[b1] (8B aligned)
            v2f A; A.x = __expf(sv.x - m); A.y = __expf(sv.y - m);
            v2f B0; B0.x = p[0];  B0.y = p[64];
            v2f B1; B1.x = p[16]; B1.y = p[80];
            v2f B2; B2.x = p[32]; B2.y = p[96];
            v2f B3; B3.x = p[48]; B3.y = p[112];
            c0 = __builtin_amdgcn_wmma_f32_16x16x4_f32(false, A, false, B0, (short)0, c0, false, false);
            c1 = __builtin_amdgcn_wmma_f32_16x16x4_f32(false, A, false, B1, (short)0, c1, false, false);
            c2 = __builtin_amdgcn_wmma_f32_16x16x4_f32(false, A, false, B2, (short)0, c2, false, false);
            c3 = __builtin_amdgcn_wmma_f32_16x16x4_f32(false, A, false, B3, (short)0, c3, false, false);
            p += 256;                             // 4 rows * 64 cols
            sp += 4;
        }
    } else {
        for (int stp = 0; stp < 128; ++stp) {
            int base = jbase + stp * 4;
            int b0 = base + (hi ? 2 : 0);
            int b1 = b0 + 1;
            int cb0 = min(b0, N - 1);
            int cb1 = min(b1, N - 1);
            float e0 = __expf(s[cb0] - m);        // unconditional exp,
            float e1 = __expf(s[cb1] - m);        // masked by select after
            v2f A; A.x = (b0 < N) ? e0 : 0.f;
                   A.y = (b1 < N) ? e1 : 0.f;
            const float* p0 = nx + (size_t)cb0 * 64 + (size_t)col;
            const float* p1 = nx + (size_t)cb1 * 64 + (size_t)col;
            v2f B0; B0.x = p0[0];  B0.y = p1[0];
            v2f B1; B1.x = p0[16]; B1.y = p1[16];
            v2f B2; B2.x = p0[32]; B2.y = p1[32];
            v2f B3; B3.x = p0[48]; B3.y = p1[48];
            c0 = __builtin_amdgcn_wmma_f32_16x16x4_f32(false, A, false, B0, (short)0, c0, false, false);
            c1 = __builtin_amdgcn_wmma_f32_16x16x4_f32(false, A, false, B1, (short)0, c1, false, false);
            c2 = __builtin_amdgcn_wmma_f32_16x16x4_f32(false, A, false, B2, (short)0, c2, false, false);
            c3 = __builtin_amdgcn_wmma_f32_16x16x4_f32(false, A, false, B3, (short)0, c3, false, false);
        }
    }

    // Row M=0 of each C tile lives in VGPR0 on lanes 0..15.
    if (!hi) {
        lds_agg[wave * 64 +  0 + col] = c0[0];
        lds_agg[wave * 64 + 16 + col] = c1[0];
        lds_agg[wave * 64 + 32 + col] = c2[0];
        lds_agg[wave * 64 + 48 + col] = c3[0];
    }
    __syncthreads();
    if (tid < 64) {
        float acc = 0.f;
#pragma unroll
        for (int w = 0; w < 8; ++w) acc += lds_agg[w * 64 + tid];
        bagg[blockIdx.x * 64 + tid] = acc;
    }
}

// ---------------------------------------------------------------------------
// Kernel 5: reduce partials, agg = (Σ bagg)/Z, out = sigmoid(W @ agg)
// ---------------------------------------------------------------------------
__global__ void k_out(const float* __restrict__ W, const float* __restrict__ bsum,
                      const float* __restrict__ bagg, int nb, float* __restrict__ out) {
    __shared__ float agg[64];
    int tid = threadIdx.x;                 // 64 threads
    float Z = 0.f;
    for (int b = 0; b < nb; ++b) Z += bsum[b];
    float a = 0.f;
    for (int b = 0; b < nb; ++b) a += bagg[b * 64 + tid];
    agg[tid] = a / Z;
    __syncthreads();
    float o = 0.f;
#pragma unroll 8
    for (int i = 0; i < 64; ++i) o += W[tid * 64 + i] * agg[i];
    out[tid] = 1.f / (1.f + __expf(-o));
}

// ---------------------------------------------------------------------------
extern "C" void kernel_launch(void* const* d_in, const int* in_sizes, int n_in,
                              void* d_out, int out_size, void* d_ws, size_t ws_size,
                              hipStream_t stream) {
    // inputs: x(64) [unused: softmax-invariant], n_x(N*64), W(64*64), a(128)
    const float* nx = (const float*)d_in[1];
    const float* W  = (const float*)d_in[2];
    const float* a  = (const float*)d_in[3];
    float* out = (float*)d_out;

    int N  = in_sizes[1] / 64;    // 500000
    int N4 = N / 4;               // N divisible by 4

    float* ws   = (float*)d_ws;
    float* v    = ws;             // 64
    float* s    = ws + 64;        // N (16B-aligned: 64 floats = 256B offset)
    float* bmax = s + N;          // up to 1024
    float* mPtr = bmax + 1024;    // 1
    float* bsum = mPtr + 1;       // up to 1024
    float* bagg = bsum + 1024;    // nbD * 64

    int nbB = (N + 4095) / 4096;
    int nbD = (N + 4095) / 4096;

    hipLaunchKernelGGL(k_fold_v, dim3(1),   dim3(64),  0, stream, W, a, v);
    hipLaunchKernelGGL(k_energy, dim3(nbB), dim3(256), 0, stream,
                       (const float4*)nx, v, (float4*)s, bmax, N4);
    hipLaunchKernelGGL(k_max,    dim3(1),   dim3(256), 0, stream, bmax, mPtr, nbB);
    hipLaunchKernelGGL(k_agg,    dim3(nbD), dim3(256), 0, stream, nx, s, mPtr, bsum, bagg, N);
    hipLaunchKernelGGL(k_out,    dim3(1),   dim3(64),  0, stream, W, bsum, bagg, nbD, out);
}